// SVDTransform_3058016715402
// MI455X (gfx1250) — compile-verified
//
#include <hip/hip_runtime.h>
#include <math.h>

// ---------------------------------------------------------------------------
// Batched Kabsch / Procrustes for MI455X (gfx1250, wave32).
// Stage 1: per-batch extended covariance via V_WMMA_F32_16X16X4_F32.
//   A (16x4 f32): rows = 4 blocks of (src_x, src_y, src_z, 1), block r uses
//                 points p0+4r .. p0+4r+3 as the K=4 window.
//   B (4x16 f32): cols = 4 blocks of (tgt_x, tgt_y, tgt_z, 1), same windows.
//   Diagonal 4x4 blocks of C accumulate  sum [src;1] * [tgt;1]^T  over 16
//   points per WMMA -> 64 WMMAs reduce one batch (N=1024) in one wave.
//   Branchless operand build (loads + v_cndmask) keeps EXEC all-1s in the
//   hot loop; two accumulators break the WMMA C->C dependency chain.
// Stage 2: per-batch 3x3 SVD (Jacobi on H^T H) -> R, t.
// ---------------------------------------------------------------------------

typedef float v2f __attribute__((ext_vector_type(2)));
typedef float v8f __attribute__((ext_vector_type(8)));

#define NPTS 1024   // points per batch (reference: (4096, 1024, 3))

// ---------------------------- Stage 1 --------------------------------------
__global__ __launch_bounds__(128) void cov_wmma_kernel(
    const float* __restrict__ src, const float* __restrict__ tgt,
    float* __restrict__ ws) {
  const int lane  = threadIdx.x & 31;
  const int wave  = threadIdx.x >> 5;
  const int batch = blockIdx.x * 4 + wave;

  // lane -> (row-block r, component c) ; ISA A-layout: lanes<16 hold K=0/1,
  // lanes>=16 hold K=2/3 (B symmetric).
  const int  r    = (lane >> 2) & 3;
  const int  comp = lane & 3;
  const bool hi   = lane >= 16;
  const bool isw  = comp < 3;          // lanes carrying x/y/z (vs the 1-row)

  const size_t base = (size_t)batch * (NPTS * 3);
  // first point this lane reads inside each 16-point window
  const int p_even = 4 * r + (hi ? 2 : 0);
  const int e0     = p_even * 3 + (isw ? comp : 0);  // comp==3: safe dup addr

  const float* ps = src + base + e0;
  const float* pt = tgt + base + e0;

  v8f acc0 = {0.f, 0.f, 0.f, 0.f, 0.f, 0.f, 0.f, 0.f};
  v8f acc1 = {0.f, 0.f, 0.f, 0.f, 0.f, 0.f, 0.f, 0.f};

#pragma unroll 4
  for (int i = 0; i < NPTS / 32; ++i) {   // 2 WMMAs per iteration
    const int off0 = (2 * i) * 48;        // 16 points * 3 floats
    const int off1 = (2 * i + 1) * 48;

    // unconditional loads; comp==3 lanes get 1.0 via cndmask (no EXEC churn)
    float a0x = ps[off0], a0y = ps[off0 + 3];
    float b0x = pt[off0], b0y = pt[off0 + 3];
    float a1x = ps[off1], a1y = ps[off1 + 3];
    float b1x = pt[off1], b1y = pt[off1 + 3];

    v2f a, b;
    a.x = isw ? a0x : 1.0f;  a.y = isw ? a0y : 1.0f;
    b.x = isw ? b0x : 1.0f;  b.y = isw ? b0y : 1.0f;
    acc0 = __builtin_amdgcn_wmma_f32_16x16x4_f32(false, a, false, b,
                                                 (short)0, acc0, false, false);

    a.x = isw ? a1x : 1.0f;  a.y = isw ? a1y : 1.0f;
    b.x = isw ? b1x : 1.0f;  b.y = isw ? b1y : 1.0f;
    acc1 = __builtin_amdgcn_wmma_f32_16x16x4_f32(false, a, false, b,
                                                 (short)0, acc1, false, false);
  }

  // ---- extract diagonal 4x4 blocks: cov[i][j] = sum_r C[4r+i][4r+j] ----
  __shared__ float E[4][16][16];
  const int col   = lane & 15;
  const int mbase = hi ? 8 : 0;   // C layout: VGPR v = row v (lanes<16) / v+8
#pragma unroll
  for (int v = 0; v < 8; ++v) E[wave][mbase + v][col] = acc0[v] + acc1[v];
  __syncthreads();

  if (lane < 16) {
    const int i = lane >> 2, j = lane & 3;
    float sum = 0.f;
#pragma unroll
    for (int rr = 0; rr < 4; ++rr) sum += E[wave][4 * rr + i][4 * rr + j];
    // ws layout per batch (16 floats):
    //  [i*4+j] i,j<3 : sum src_i*tgt_j ; [i*4+3]: sum src_i ;
    //  [12+j]        : sum tgt_j       ; [15]   : N
    ws[(size_t)batch * 16 + lane] = sum;
  }
}

// ---------------------------- Stage 2 --------------------------------------
__device__ __forceinline__ void jrot(float B[3][3], float V[3][3], int p, int q) {
  const float apq = B[p][q];
  const float app = B[p][p], aqq = B[q][q];
  if (fabsf(apq) <= 1e-12f * (fabsf(app) + fabsf(aqq))) return;
  const float tau = (aqq - app) / (2.0f * apq);
  const float t   = (tau >= 0.0f ? 1.0f : -1.0f) /
                    (fabsf(tau) + sqrtf(1.0f + tau * tau));
  const float c = 1.0f / sqrtf(1.0f + t * t);
  const float s = t * c;
#pragma unroll
  for (int k = 0; k < 3; ++k) {
    const float bkp = B[k][p], bkq = B[k][q];
    B[k][p] = c * bkp - s * bkq;
    B[k][q] = s * bkp + c * bkq;
  }
#pragma unroll
  for (int k = 0; k < 3; ++k) {
    const float bpk = B[p][k], bqk = B[q][k];
    B[p][k] = c * bpk - s * bqk;
    B[q][k] = s * bpk + c * bqk;
  }
#pragma unroll
  for (int k = 0; k < 3; ++k) {
    const float vkp = V[k][p], vkq = V[k][q];
    V[k][p] = c * vkp - s * vkq;
    V[k][q] = s * vkp + c * vkq;
  }
}

__device__ __forceinline__ void cswap(float lam[3], float V[3][3], int a, int b) {
  if (lam[a] < lam[b]) {
    const float tl = lam[a]; lam[a] = lam[b]; lam[b] = tl;
#pragma unroll
    for (int k = 0; k < 3; ++k) {
      const float tv = V[k][a]; V[k][a] = V[k][b]; V[k][b] = tv;
    }
  }
}

__global__ __launch_bounds__(256) void svd_finalize_kernel(
    const float* __restrict__ ws, float* __restrict__ outR,
    float* __restrict__ outT, int bs) {
  const int b = blockIdx.x * 256 + threadIdx.x;
  if (b >= bs) return;

  float e[16];
#pragma unroll
  for (int k = 0; k < 16; ++k) e[k] = ws[(size_t)b * 16 + k];

  const float inv_n = 1.0f / e[15];
  float sm[3], tm[3], H[3][3];
#pragma unroll
  for (int i = 0; i < 3; ++i) sm[i] = e[i * 4 + 3] * inv_n;
#pragma unroll
  for (int j = 0; j < 3; ++j) tm[j] = e[12 + j] * inv_n;
#pragma unroll
  for (int i = 0; i < 3; ++i)
#pragma unroll
    for (int j = 0; j < 3; ++j)
      H[i][j] = e[i * 4 + j] - e[i * 4 + 3] * e[12 + j] * inv_n;

  // B = H^T H (symmetric), Jacobi eigendecomposition -> V (right sing. vecs)
  float B[3][3];
#pragma unroll
  for (int a = 0; a < 3; ++a)
#pragma unroll
    for (int c = 0; c < 3; ++c)
      B[a][c] = H[0][a] * H[0][c] + H[1][a] * H[1][c] + H[2][a] * H[2][c];

  float V[3][3] = {{1.f, 0.f, 0.f}, {0.f, 1.f, 0.f}, {0.f, 0.f, 1.f}};
#pragma unroll
  for (int sweep = 0; sweep < 6; ++sweep) {
    jrot(B, V, 0, 1);
    jrot(B, V, 0, 2);
    jrot(B, V, 1, 2);
  }

  float lam[3] = {B[0][0], B[1][1], B[2][2]};
  cswap(lam, V, 0, 1); cswap(lam, V, 1, 2); cswap(lam, V, 0, 1);  // desc

  // U columns: u_k = H v_k / sigma_k
  float U[3][3], sig[3];
#pragma unroll
  for (int k = 0; k < 3; ++k) {
    const float w0 = H[0][0] * V[0][k] + H[0][1] * V[1][k] + H[0][2] * V[2][k];
    const float w1 = H[1][0] * V[0][k] + H[1][1] * V[1][k] + H[1][2] * V[2][k];
    const float w2 = H[2][0] * V[0][k] + H[2][1] * V[1][k] + H[2][2] * V[2][k];
    const float nn = sqrtf(w0 * w0 + w1 * w1 + w2 * w2);
    sig[k] = nn;
    const float inv = 1.0f / fmaxf(nn, 1e-30f);
    U[0][k] = w0 * inv; U[1][k] = w1 * inv; U[2][k] = w2 * inv;
  }
  if (sig[2] < 1e-5f * sig[0] + 1e-30f) {  // degenerate: u2 = u0 x u1
    U[0][2] = U[1][0] * U[2][1] - U[2][0] * U[1][1];
    U[1][2] = U[2][0] * U[0][1] - U[0][0] * U[2][1];
    U[2][2] = U[0][0] * U[1][1] - U[1][0] * U[0][1];
  }

  // R0 = V U^T ; det fix negates V's last (smallest-sigma) column
  float R0[3][3];
#pragma unroll
  for (int i = 0; i < 3; ++i)
#pragma unroll
    for (int j = 0; j < 3; ++j)
      R0[i][j] = V[i][0] * U[j][0] + V[i][1] * U[j][1] + V[i][2] * U[j][2];
  const float det =
      R0[0][0] * (R0[1][1] * R0[2][2] - R0[1][2] * R0[2][1]) -
      R0[0][1] * (R0[1][0] * R0[2][2] - R0[1][2] * R0[2][0]) +
      R0[0][2] * (R0[1][0] * R0[2][1] - R0[1][1] * R0[2][0]);
  const float sgn = (det < 0.0f) ? -1.0f : 1.0f;

  float R[3][3];
#pragma unroll
  for (int i = 0; i < 3; ++i)
#pragma unroll
    for (int j = 0; j < 3; ++j)
      R[i][j] = V[i][0] * U[j][0] + V[i][1] * U[j][1] + sgn * V[i][2] * U[j][2];

#pragma unroll
  for (int i = 0; i < 3; ++i) {
#pragma unroll
    for (int j = 0; j < 3; ++j) outR[(size_t)b * 9 + i * 3 + j] = R[i][j];
    outT[(size_t)b * 3 + i] =
        tm[i] - (R[i][0] * sm[0] + R[i][1] * sm[1] + R[i][2] * sm[2]);
  }
}

// ---------------------------- launcher -------------------------------------
extern "C" void kernel_launch(void* const* d_in, const int* in_sizes, int n_in,
                              void* d_out, int out_size, void* d_ws,
                              size_t ws_size, hipStream_t stream) {
  const float* src = (const float*)d_in[0];
  const float* tgt = (const float*)d_in[1];
  float* out = (float*)d_out;
  const int bs = in_sizes[0] / (NPTS * 3);  // 4096

  float* ws = (float*)d_ws;  // bs*16 floats, fully overwritten each call

  cov_wmma_kernel<<<bs / 4, 128, 0, stream>>>(src, tgt, ws);
  svd_finalize_kernel<<<(bs + 255) / 256, 256, 0, stream>>>(
      ws, out, out + (size_t)bs * 9, bs);
}